// Csolver_59442347377031
// MI455X (gfx1250) — compile-verified
//
#include <hip/hip_runtime.h>

// Batched complex solve (A_r + iA_i) x = (b_r + i b_i), B=512, N=128, K=16.
// One workgroup per batch. Augmented [A|b] lives in LDS (2 fp32 planes,
// 128 x 145 pitch). Blocked LU (NB=16) with partial pivoting; trailing
// submatrix updates (the O(N^3) term) run on v_wmma_f32_16x16x4_f32.

typedef float v2f __attribute__((ext_vector_type(2)));
typedef float v8f __attribute__((ext_vector_type(8)));

#define NN      128        // matrix dim
#define KK      16         // rhs count
#define WID     144        // augmented width (A cols + rhs cols)
#define P       145        // LDS pitch (odd -> conflict-free column access)
#define NTHR    256        // 8 waves of 32
#define NB      16         // LU block size

__global__ __launch_bounds__(NTHR, 1)
void csolve_lu_wmma(const float* __restrict__ gAr, const float* __restrict__ gAi,
                    const float* __restrict__ gBr, const float* __restrict__ gBi,
                    float* __restrict__ out, int batches)
{
    extern __shared__ float smem[];
    float* Mr = smem;                 // [128][145] real plane
    float* Mi = Mr + NN * P;          // [128][145] imag plane
    float* rv = Mi + NN * P;          // [128] argmax values
    int*   ri = (int*)(rv + NN);      // [128] argmax indices

    const int tid  = threadIdx.x;
    const int bidx = blockIdx.x;
    const int lane = tid & 31;
    const int wv   = tid >> 5;
    const int lm   = lane & 15;       // M (A frag) / N (B,C frags)
    const int hi   = lane >> 4;       // lane half

    const float* ar = gAr + (size_t)bidx * NN * NN;
    const float* ai = gAi + (size_t)bidx * NN * NN;
    const float* br = gBr + (size_t)bidx * NN * KK;
    const float* bi = gBi + (size_t)bidx * NN * KK;

    // ---- Load A (cols 0..127) and b (cols 128..143) into LDS, float4 global reads
    for (int q = tid; q < NN * NN / 4; q += NTHR) {
        int r = q >> 5;             // q*4 / 128
        int c = (q & 31) << 2;
        float4 vr = ((const float4*)ar)[q];
        float4 vi = ((const float4*)ai)[q];
        Mr[r*P + c+0] = vr.x; Mr[r*P + c+1] = vr.y; Mr[r*P + c+2] = vr.z; Mr[r*P + c+3] = vr.w;
        Mi[r*P + c+0] = vi.x; Mi[r*P + c+1] = vi.y; Mi[r*P + c+2] = vi.z; Mi[r*P + c+3] = vi.w;
    }
    for (int q = tid; q < NN * KK / 4; q += NTHR) {
        int r = q >> 2;             // q*4 / 16
        int c = (q & 3) << 2;
        float4 vr = ((const float4*)br)[q];
        float4 vi = ((const float4*)bi)[q];
        Mr[r*P + NN + c+0] = vr.x; Mr[r*P + NN + c+1] = vr.y;
        Mr[r*P + NN + c+2] = vr.z; Mr[r*P + NN + c+3] = vr.w;
        Mi[r*P + NN + c+0] = vi.x; Mi[r*P + NN + c+1] = vi.y;
        Mi[r*P + NN + c+2] = vi.z; Mi[r*P + NN + c+3] = vi.w;
    }
    __syncthreads();

    // ---- Blocked LU with partial pivoting over augmented matrix
    for (int kb = 0; kb < NN / NB; ++kb) {
        const int c0 = kb * NB;

        // Panel factorization: columns c0 .. c0+15
        for (int jj = 0; jj < NB; ++jj) {
            const int c = c0 + jj;

            // argmax |M[r][c]|^2 over r in [c,128)
            if (tid < NN) {
                float v = -1.0f; int idx = c;
                if (tid >= c) {
                    float xr = Mr[tid*P + c], xi = Mi[tid*P + c];
                    v = xr*xr + xi*xi; idx = tid;
                }
                rv[tid] = v; ri[tid] = idx;
            }
            __syncthreads();
            for (int s = NN >> 1; s > 0; s >>= 1) {
                if (tid < s) {
                    if (rv[tid + s] > rv[tid]) { rv[tid] = rv[tid + s]; ri[tid] = ri[tid + s]; }
                }
                __syncthreads();
            }
            const int piv = ri[0];

            // full-row swap (width 144)
            if (piv != c) {
                for (int cc = tid; cc < WID; cc += NTHR) {
                    float tr = Mr[c*P + cc]; Mr[c*P + cc] = Mr[piv*P + cc]; Mr[piv*P + cc] = tr;
                    float ti = Mi[c*P + cc]; Mi[c*P + cc] = Mi[piv*P + cc]; Mi[piv*P + cc] = ti;
                }
            }
            __syncthreads();

            // scale subdiagonal column by 1/pivot (complex reciprocal)
            {
                float pr = Mr[c*P + c], pi = Mi[c*P + c];
                float den = pr*pr + pi*pi;
                float irc = pr / den, iic = -pi / den;
                if (tid > c && tid < NN) {
                    float xr = Mr[tid*P + c], xi = Mi[tid*P + c];
                    Mr[tid*P + c] = xr*irc - xi*iic;
                    Mi[tid*P + c] = xr*iic + xi*irc;
                }
            }
            __syncthreads();

            // rank-1 update restricted to remaining panel columns
            const int ncc = (c0 + NB) - (c + 1);
            if (ncc > 0) {
                const int tot = (NN - (c + 1)) * ncc;
                for (int w = tid; w < tot; w += NTHR) {
                    int r  = c + 1 + w / ncc;
                    int cc = c + 1 + w % ncc;
                    float lr = Mr[r*P + c],  li = Mi[r*P + c];
                    float ur = Mr[c*P + cc], ui = Mi[c*P + cc];
                    Mr[r*P + cc] -= lr*ur - li*ui;
                    Mi[r*P + cc] -= lr*ui + li*ur;
                }
            }
            __syncthreads();
        }

        // Block-row update: U12 = L11^{-1} * A12 (unit-lower tri solve), col-parallel
        const int r0 = c0 + NB;
        for (int cc = r0 + tid; cc < WID; cc += NTHR) {
            for (int j = 1; j < NB; ++j) {
                int r = c0 + j;
                float sr = Mr[r*P + cc], si = Mi[r*P + cc];
                for (int m = 0; m < j; ++m) {
                    float lr = Mr[r*P + c0 + m],  li = Mi[r*P + c0 + m];
                    float ur = Mr[(c0+m)*P + cc], ui = Mi[(c0+m)*P + cc];
                    sr -= lr*ur - li*ui;
                    si -= lr*ui + li*ur;
                }
                Mr[r*P + cc] = sr; Mi[r*P + cc] = si;
            }
        }
        __syncthreads();

        // Trailing update: A22 -= L21 * U12 (complex) via WMMA f32 16x16x4 chains
        const int mtiles = (NN  - r0) >> 4;
        const int ntiles = (WID - r0) >> 4;
        const int ntot   = mtiles * ntiles;
        for (int t = wv; t < ntot; t += NTHR / 32) {
            const int rowA = r0 + (t / ntiles) * 16;   // C tile rows
            const int colC = r0 + (t % ntiles) * 16;   // C tile cols
            v8f cr = {};
            v8f ci = {};
            for (int kk = 0; kk < 4; ++kk) {
                const int kbase = c0 + kk * 4 + 2 * hi;   // K = vgpr + 2*half
                // A fragment (L21): lane lm = M row, two K values
                v2f far, fai, fain;
                far.x = Mr[(rowA + lm)*P + kbase];     far.y = Mr[(rowA + lm)*P + kbase + 1];
                fai.x = Mi[(rowA + lm)*P + kbase];     fai.y = Mi[(rowA + lm)*P + kbase + 1];
                fain.x = -fai.x;                       fain.y = -fai.y;
                // B fragment (U12): lane lm = N col, two K values (rows of panel)
                v2f fbr, fbi;
                fbr.x = Mr[(kbase)*P + colC + lm];     fbr.y = Mr[(kbase + 1)*P + colC + lm];
                fbi.x = Mi[(kbase)*P + colC + lm];     fbi.y = Mi[(kbase + 1)*P + colC + lm];
                // cr += Ar*Br - Ai*Bi ; ci += Ar*Bi + Ai*Br
                cr = __builtin_amdgcn_wmma_f32_16x16x4_f32(false, far,  false, fbr, (short)0, cr, false, false);
                cr = __builtin_amdgcn_wmma_f32_16x16x4_f32(false, fain, false, fbi, (short)0, cr, false, false);
                ci = __builtin_amdgcn_wmma_f32_16x16x4_f32(false, far,  false, fbi, (short)0, ci, false, false);
                ci = __builtin_amdgcn_wmma_f32_16x16x4_f32(false, fai,  false, fbr, (short)0, ci, false, false);
            }
            // C layout: vgpr r -> M = r + 8*half, N = lane&15 ; subtract product
            for (int r = 0; r < 8; ++r) {
                int gr = rowA + r + 8 * hi;
                int gc = colC + lm;
                Mr[gr*P + gc] -= cr[r];
                Mi[gr*P + gc] -= ci[r];
            }
        }
        __syncthreads();
    }

    // ---- Back substitution: U x = y (y sits in cols 128..143)
    for (int r = NN - 1; r >= 0; --r) {
        {
            float pr = Mr[r*P + r], pi = Mi[r*P + r];
            float den = pr*pr + pi*pi;
            float irc = pr / den, iic = -pi / den;
            if (tid < KK) {
                float yr = Mr[r*P + NN + tid], yi = Mi[r*P + NN + tid];
                Mr[r*P + NN + tid] = yr*irc - yi*iic;
                Mi[r*P + NN + tid] = yr*iic + yi*irc;
            }
        }
        __syncthreads();
        const int tot = r * KK;
        for (int w = tid; w < tot; w += NTHR) {
            int i = w >> 4, k = w & 15;
            float ur = Mr[i*P + r], ui = Mi[i*P + r];
            float xr = Mr[r*P + NN + k], xi = Mi[r*P + NN + k];
            Mr[i*P + NN + k] -= ur*xr - ui*xi;
            Mi[i*P + NN + k] -= ur*xi + ui*xr;
        }
        __syncthreads();
    }

    // ---- Write out: real plane then imag plane, [B][N][K] each
    float* outR = out;
    float* outI = out + (size_t)batches * NN * KK;
    const size_t obase = (size_t)bidx * NN * KK;
    for (int q = tid; q < NN * KK; q += NTHR) {
        int r = q >> 4, k = q & 15;
        outR[obase + q] = Mr[r*P + NN + k];
        outI[obase + q] = Mi[r*P + NN + k];
    }
}

extern "C" void kernel_launch(void* const* d_in, const int* in_sizes, int n_in,
                              void* d_out, int out_size, void* d_ws, size_t ws_size,
                              hipStream_t stream) {
    const float* Ar = (const float*)d_in[0];
    const float* Ai = (const float*)d_in[1];
    const float* Br = (const float*)d_in[2];
    const float* Bi = (const float*)d_in[3];
    float* out = (float*)d_out;

    const int batches = in_sizes[0] / (NN * NN);   // 512
    const size_t shmem = (size_t)(2 * NN * P) * sizeof(float)   // two planes
                       + (size_t)NN * sizeof(float)             // argmax vals
                       + (size_t)NN * sizeof(int);              // argmax idx

    csolve_lu_wmma<<<batches, NTHR, shmem, stream>>>(Ar, Ai, Br, Bi, out, batches);
}